// BotGraphSAGE_80573586473705
// MI455X (gfx1250) — compile-verified
//
#include <hip/hip_runtime.h>

// ---------------------------------------------------------------------------
// BotGraphSAGE forward on gfx1250 (MI455X).
// Dense layers -> V_WMMA_F32_16X16X4_F32, fp32 (matches fp32 reference):
//   * compile-time shapes -> immediate-offset global_load_b64 A fragments
//   * W staged transposed in LDS (stride 68 -> conflict-free ds_load_b64)
//   * one wave computes a 16x64 C block (4 accumulators, A reused 4x)
// SAGE aggregation -> coalesced gather + global_atomic_add_f32 scatter-mean.
// ---------------------------------------------------------------------------

#define NN 50000
#define TILES_M 3125   // 50000 / 16, exact

typedef __attribute__((ext_vector_type(2))) float v2f;
typedef __attribute__((ext_vector_type(8))) float v8f;

enum { ACT_NONE = 0, ACT_LEAKY = 1, ACT_RELU = 2 };

__device__ __forceinline__ v8f wmma_f32_k4(v2f a, v2f b, v8f c) {
  // 8 args: (neg_a, A, neg_b, B, c_mod, C, reuse_a, reuse_b)
  return __builtin_amdgcn_wmma_f32_16x16x4_f32(false, a, false, b, (short)0, c,
                                               false, false);
}

template <int N> struct Log2 { static constexpr int v = 1 + Log2<N / 2>::v; };
template <> struct Log2<1> { static constexpr int v = 0; };

constexpr int KBLK = 64;        // K-chunk staged in LDS at a time
constexpr int KPB  = KBLK + 4;  // 68: LDS column stride -> banks 4*l16+2*half,
                                // all 32 lanes' b64 reads hit distinct banks

// Accumulate acc[0..NT) += A[16 x K] @ W[K x NACT] for this wave's tile.
// W is staged transposed through LDS in KBLK chunks (all 256 threads help).
template <int K, int LDA, int NACT, int NT>
__device__ __forceinline__ void gemm_accum(const float* __restrict__ A,
                                           const float* __restrict__ W,
                                           float* __restrict__ sW, bool valid,
                                           int rowA, int half,
                                           const int* colv, const float* msk,
                                           v8f* acc) {
  const int tid = threadIdx.x;
  const float* __restrict__ Arow = A + (size_t)rowA * LDA + 2 * half;
  constexpr int L2N = Log2<NACT>::v;
  #pragma unroll
  for (int k0 = 0; k0 < K; k0 += KBLK) {
    const int kb = (K - k0) < KBLK ? (K - k0) : KBLK;  // folds after unroll
    __syncthreads();                                    // chunk reuse fence
    // stage W[k0:k0+kb, :] transposed: sW[c*KPB + k] = W[(k0+k)*NACT + c]
    for (int i = tid; i < NACT * kb; i += 256) {
      const int c = i & (NACT - 1);
      const int k = i >> L2N;
      sW[c * KPB + k] = W[k0 * NACT + i];               // contiguous global read
    }
    __syncthreads();
    if (valid) {
      #pragma unroll 16
      for (int k = 0; k < kb; k += 4) {
        const v2f a = *(const v2f*)(Arow + k0 + k);     // imm-offset b64
        #pragma unroll
        for (int t = 0; t < NT; ++t) {
          v2f b = *(const v2f*)(sW + colv[t] * KPB + k + 2 * half);
          if constexpr ((NACT & 15) != 0) { b.x *= msk[t]; b.y *= msk[t]; }
          acc[t] = wmma_f32_k4(a, b, acc[t]);
        }
      }
    }
  }
}

// C[50000 x NACT] = act( A1 @ W1 (+ A2 @ W2) + bias ), one 16x(16*NT) block / wave
template <int K1, int LDA1, bool HAS2, int K2, int LDA2, int NACT, int LDC,
          int ACT, int NT>
__global__ __launch_bounds__(256) void gemm_t(const float* __restrict__ A1,
                                              const float* __restrict__ W1,
                                              const float* __restrict__ A2,
                                              const float* __restrict__ W2,
                                              const float* __restrict__ bias,
                                              float* __restrict__ C) {
  constexpr int TN = (NACT + 15) / 16;
  constexpr int GROUPS = (TN + NT - 1) / NT;
  __shared__ float sW[NACT * KPB];

  const int tid  = threadIdx.x;
  const int lane = tid & 31;
  const int wave = tid >> 5;
  const int tile = blockIdx.x * 8 + wave;
  const bool valid = tile < TILES_M * GROUPS;   // wave-uniform
  const int tt = valid ? tile : 0;
  const int rowTile  = tt / GROUPS;
  const int colGroup = tt - rowTile * GROUPS;
  const int half = lane >> 4;                   // 0: lanes 0-15, 1: lanes 16-31
  const int l16  = lane & 15;
  const int rowA = rowTile * 16 + l16;

  int colv[NT];
  float msk[NT];
  #pragma unroll
  for (int t = 0; t < NT; ++t) {
    const int c = (colGroup * NT + t) * 16 + l16;
    msk[t]  = (c < NACT) ? 1.f : 0.f;
    colv[t] = (c < NACT) ? c : 0;
  }

  v8f acc[NT];
  const v8f z = {};
  #pragma unroll
  for (int t = 0; t < NT; ++t) acc[t] = z;

  gemm_accum<K1, LDA1, NACT, NT>(A1, W1, sW, valid, rowA, half, colv, msk, acc);
  if constexpr (HAS2)
    gemm_accum<K2, LDA2, NACT, NT>(A2, W2, sW, valid, rowA, half, colv, msk, acc);

  if (!valid) return;
  #pragma unroll
  for (int t = 0; t < NT; ++t) {
    const int c = (colGroup * NT + t) * 16 + l16;
    const float bv = bias[colv[t]];
    #pragma unroll
    for (int r = 0; r < 8; ++r) {
      const int row = rowTile * 16 + r + 8 * half;   // C layout: vgpr r, half
      float v = acc[t][r] + bv;
      if constexpr (ACT == ACT_LEAKY) v = v > 0.f ? v : 0.01f * v;
      if constexpr (ACT == ACT_RELU)  v = fmaxf(v, 0.f);
      if ((NACT & 15) == 0 || c < NACT) C[(size_t)row * LDC + c] = v;
    }
  }
}

template <int K1, int LDA1, bool HAS2, int K2, int LDA2, int NACT, int LDC,
          int ACT, int NT>
static inline void launch_gemm(hipStream_t stream, const float* A1,
                               const float* W1, const float* A2,
                               const float* W2, const float* bias, float* C) {
  constexpr int TN = (NACT + 15) / 16;
  constexpr int GROUPS = (TN + NT - 1) / NT;
  constexpr int blocks = (TILES_M * GROUPS + 7) / 8;  // 8 waves per block
  hipLaunchKernelGGL((gemm_t<K1, LDA1, HAS2, K2, LDA2, NACT, LDC, ACT, NT>),
                     dim3(blocks), dim3(256), 0, stream, A1, W1, A2, W2, bias, C);
}

// ---------------------------------------------------------------------------

// LeakyReLU(num@W_num+b) -> cols [32,74); LeakyReLU(cat@W_cat+b) -> [74,116)
__global__ void nc_kernel(const float* __restrict__ num,
                          const float* __restrict__ cat,
                          const float* __restrict__ Wn,
                          const float* __restrict__ bn,
                          const float* __restrict__ Wc,
                          const float* __restrict__ bc,
                          float* __restrict__ x116) {
  int idx = blockIdx.x * blockDim.x + threadIdx.x;
  if (idx >= NN * 84) return;
  int node = idx / 84;
  int j = idx - node * 84;
  float v;
  int col;
  if (j < 42) {
    const float* r = num + (size_t)node * 4;
    v = bn[j] + r[0] * Wn[0 * 42 + j] + r[1] * Wn[1 * 42 + j] +
        r[2] * Wn[2 * 42 + j] + r[3] * Wn[3 * 42 + j];
    col = 32 + j;
  } else {
    j -= 42;
    const float* r = cat + (size_t)node * 3;
    v = bc[j] + r[0] * Wc[j] + r[1] * Wc[42 + j] + r[2] * Wc[84 + j];
    col = 74 + j;
  }
  x116[(size_t)node * 116 + col] = v > 0.f ? v : 0.01f * v;
}

__global__ void fill0_kernel(float* __restrict__ p, int n) {
  int i = blockIdx.x * blockDim.x + threadIdx.x;
  if (i < n) p[i] = 0.f;
}

__global__ void count_kernel(const int* __restrict__ dst, int E,
                             float* __restrict__ cnt) {
  int e = blockIdx.x * blockDim.x + threadIdx.x;
  if (e >= E) return;
  __hip_atomic_fetch_add(&cnt[dst[e]], 1.0f, __ATOMIC_RELAXED,
                         __HIP_MEMORY_SCOPE_AGENT);
}

// one thread per (edge, 4-float chunk); an edge's chunk threads are adjacent
// -> coalesced D*4-byte gather of x[src] row, atomics hit one agg row.
template <int D>
__global__ void scatter_kernel(const int* __restrict__ src,
                               const int* __restrict__ dst, int E,
                               const float* __restrict__ x,
                               float* __restrict__ agg) {
  constexpr int CH = D / 4;
  int idx = blockIdx.x * blockDim.x + threadIdx.x;
  if (idx >= E * CH) return;
  int e  = idx / CH;                       // CH is a power of two -> shift
  int c4 = (idx - e * CH) << 2;
  int s = src[e], t = dst[e];
  const float4 v = *(const float4*)(x + (size_t)s * D + c4);
  float* o = agg + (size_t)t * D + c4;
  __hip_atomic_fetch_add(o + 0, v.x, __ATOMIC_RELAXED, __HIP_MEMORY_SCOPE_AGENT);
  __hip_atomic_fetch_add(o + 1, v.y, __ATOMIC_RELAXED, __HIP_MEMORY_SCOPE_AGENT);
  __hip_atomic_fetch_add(o + 2, v.z, __ATOMIC_RELAXED, __HIP_MEMORY_SCOPE_AGENT);
  __hip_atomic_fetch_add(o + 3, v.w, __ATOMIC_RELAXED, __HIP_MEMORY_SCOPE_AGENT);
}

template <int D>
__global__ void mean_kernel(float* __restrict__ agg,
                            const float* __restrict__ cnt, int total) {
  int i = blockIdx.x * blockDim.x + threadIdx.x;
  if (i >= total) return;
  agg[i] /= fmaxf(cnt[i / D], 1.0f);
}

// One SAGE conv: out = relu(mean_agg(xin) @ Wl + xin @ Wr + bl)
template <int DIN, int DOUT>
static void conv_launch(hipStream_t stream, const float* xin, const float* Wl,
                        const float* bl, const float* Wr, float* out,
                        float* agg, float* cnt, const int* src, const int* dst,
                        int E) {
  const int na = NN * DIN;
  hipLaunchKernelGGL(fill0_kernel, dim3((na + 255) / 256), dim3(256), 0, stream, agg, na);
  hipLaunchKernelGGL(fill0_kernel, dim3((NN + 255) / 256), dim3(256), 0, stream, cnt, NN);
  hipLaunchKernelGGL(count_kernel, dim3((E + 255) / 256), dim3(256), 0, stream, dst, E, cnt);
  const int total = E * (DIN / 4);
  hipLaunchKernelGGL((scatter_kernel<DIN>), dim3((total + 255) / 256), dim3(256), 0, stream,
                     src, dst, E, xin, agg);
  hipLaunchKernelGGL((mean_kernel<DIN>), dim3((na + 255) / 256), dim3(256), 0, stream,
                     agg, cnt, na);
  launch_gemm<DIN, DIN, true, DIN, DIN, DOUT, DOUT, ACT_RELU, 4>(
      stream, agg, Wl, xin, Wr, bl, out);
}

// ---------------------------------------------------------------------------

extern "C" void kernel_launch(void* const* d_in, const int* in_sizes, int n_in,
                              void* d_out, int out_size, void* d_ws,
                              size_t ws_size, hipStream_t stream) {
  const float* des      = (const float*)d_in[0];
  const float* num_prop = (const float*)d_in[2];
  const float* cat_prop = (const float*)d_in[3];
  const int*   edge     = (const int*)d_in[4];
  const int    E        = in_sizes[4] / 2;
  const int*   src      = edge;
  const int*   dst      = edge + E;
  const float* W_des = (const float*)d_in[5],  *b_des = (const float*)d_in[6];
  const float* W_num = (const float*)d_in[7],  *b_num = (const float*)d_in[8];
  const float* W_cat = (const float*)d_in[9],  *b_cat = (const float*)d_in[10];
  const float* W_in  = (const float*)d_in[11], *b_in  = (const float*)d_in[12];
  const float* s1a_Wl = (const float*)d_in[13], *s1a_bl = (const float*)d_in[14], *s1a_Wr = (const float*)d_in[15];
  const float* s1b_Wl = (const float*)d_in[16], *s1b_bl = (const float*)d_in[17], *s1b_Wr = (const float*)d_in[18];
  const float* s2a_Wl = (const float*)d_in[19], *s2a_bl = (const float*)d_in[20], *s2a_Wr = (const float*)d_in[21];
  const float* s2b_Wl = (const float*)d_in[22], *s2b_bl = (const float*)d_in[23], *s2b_Wr = (const float*)d_in[24];
  const float* W_o1 = (const float*)d_in[25], *b_o1 = (const float*)d_in[26];
  const float* W_o2 = (const float*)d_in[27], *b_o2 = (const float*)d_in[28];

  float* ws   = (float*)d_ws;
  float* x_a  = ws;                        // [NN,128]
  float* x_b  = x_a + (size_t)NN * 128;    // [NN,128]  (x116 overlaps here)
  float* h    = x_b + (size_t)NN * 128;    // [NN,64]
  float* agg  = h   + (size_t)NN * 64;     // [NN,128]
  float* cnt  = agg + (size_t)NN * 128;    // [NN]
  float* x116 = x_b;                       // stride 116, dead once x_a is built

  // 1) d = leaky(des @ W_des + b_des) -> x116[:, 0:32]   (K=768, N=32)
  launch_gemm<768, 768, false, 4, 4, 32, 116, ACT_LEAKY, 2>(
      stream, des, W_des, nullptr, nullptr, b_des, x116);
  // 2) n,c projections (K=4 / K=3, VALU) -> x116[:, 32:116]
  hipLaunchKernelGGL(nc_kernel, dim3((NN * 84 + 255) / 256), dim3(256), 0, stream,
                     num_prop, cat_prop, W_num, b_num, W_cat, b_cat, x116);
  // 3) x = leaky(x116 @ W_in + b_in)                    (K=116, N=128)
  launch_gemm<116, 116, false, 4, 4, 128, 128, ACT_LEAKY, 4>(
      stream, x116, W_in, nullptr, nullptr, b_in, x_a);
  // 4) GraphSAGE #1
  conv_launch<128, 64>(stream, x_a, s1a_Wl, s1a_bl, s1a_Wr, h, agg, cnt, src, dst, E);
  conv_launch<64, 128>(stream, h, s1b_Wl, s1b_bl, s1b_Wr, x_b, agg, cnt, src, dst, E);
  // 5) GraphSAGE #2
  conv_launch<128, 64>(stream, x_b, s2a_Wl, s2a_bl, s2a_Wr, h, agg, cnt, src, dst, E);
  conv_launch<64, 128>(stream, h, s2b_Wl, s2b_bl, s2b_Wr, x_a, agg, cnt, src, dst, E);
  // 6) head
  launch_gemm<128, 128, false, 4, 4, 128, 128, ACT_LEAKY, 4>(
      stream, x_a, W_o1, nullptr, nullptr, b_o1, x_b);
  launch_gemm<128, 128, false, 4, 4, 2, 2, ACT_NONE, 1>(
      stream, x_b, W_o2, nullptr, nullptr, b_o2, (float*)d_out);
}